// StaticLoss_9466107921226
// MI455X (gfx1250) — compile-verified
//
#include <hip/hip_runtime.h>
#include <cstdint>

#define NUM_C   19
#define PLANE   (512 * 512)          // H*W = 2^18
#define NPIX    (8 * 512 * 512)      // B*H*W
#define PPT     4                    // pixels per thread
#define BLOCK   256
#define GRID    (NPIX / (BLOCK * PPT))   // 2048
#define CELLS   64                   // striped global accumulator cells

typedef float v4f __attribute__((ext_vector_type(4)));
typedef int   v4i __attribute__((ext_vector_type(4)));

#if __has_builtin(__builtin_amdgcn_global_load_async_to_lds_b128)
#define HAVE_ASYNC_LDS 1
typedef __attribute__((address_space(1))) v4i* glob_v4i_p;
typedef __attribute__((address_space(3))) v4i* lds_v4i_p;
#else
#define HAVE_ASYNC_LDS 0
#endif

__global__ void zero_ws_kernel(float* __restrict__ ws) {
    ws[threadIdx.x] = 0.0f;          // 2*CELLS = 128 threads
}

__global__ __launch_bounds__(BLOCK) void focal_loss_kernel(
        const float* __restrict__ in, const long long* __restrict__ tgt,
        const float* __restrict__ w, float* __restrict__ ws) {
    __shared__ float wl[NUM_C];
    __shared__ float red[2];
#if HAVE_ASYNC_LDS
    __shared__ __align__(16) long long tl[BLOCK * PPT];
#endif
    const int tid = threadIdx.x;
    const int n0  = (blockIdx.x * BLOCK + tid) * PPT;

#if HAVE_ASYNC_LDS
    // CDNA5 async global->LDS staging of the int64 targets (ASYNCcnt path).
    // The immediate offset is applied to BOTH the global and LDS addresses,
    // so one base pair covers both 16-byte halves of this thread's 4 targets.
    {
        glob_v4i_p g = (glob_v4i_p)(tgt + n0);
        lds_v4i_p  l = (lds_v4i_p)(&tl[tid * PPT]);
        __builtin_amdgcn_global_load_async_to_lds_b128(g, l, 0,  0);
        __builtin_amdgcn_global_load_async_to_lds_b128(g, l, 16, 0);
    }
#endif

    if (tid < NUM_C) wl[tid] = w[tid];
    if (tid == 0) { red[0] = 0.0f; red[1] = 0.0f; }

    // Stream all 19 channel rows for this thread's 4 pixels into registers.
    const int b  = n0 >> 18;             // PLANE = 2^18
    const int hw = n0 & (PLANE - 1);
    const v4f* in4 = (const v4f*)in;
    const size_t base4 = ((size_t)b * NUM_C * PLANE + (size_t)hw) >> 2;

    v4f xv[NUM_C];
#pragma unroll
    for (int c = 0; c < NUM_C; ++c)
        xv[c] = in4[base4 + (size_t)c * (PLANE / 4)];

    __syncthreads();                     // wl / red visible
#if HAVE_ASYNC_LDS
#if __has_builtin(__builtin_amdgcn_s_wait_asynccnt)
    __builtin_amdgcn_s_wait_asynccnt(0);
#else
    asm volatile("s_wait_asynccnt 0" ::: "memory");
#endif
#endif

    float sumL = 0.0f, sumV = 0.0f;
#pragma unroll
    for (int j = 0; j < PPT; ++j) {
#if HAVE_ASYNC_LDS
        const int t = (int)tl[tid * PPT + j];
#else
        const int t = (int)tgt[n0 + j];
#endif
        const bool valid = (t != 255);
        const int  ts    = valid ? t : 0;

        float m = xv[0][j];
#pragma unroll
        for (int c = 1; c < NUM_C; ++c) m = fmaxf(m, xv[c][j]);

        float s = 0.0f, et = 0.0f;
#pragma unroll
        for (int c = 0; c < NUM_C; ++c) {
            const float e = __expf(xv[c][j] - m);
            s += e;
            et = (c == ts) ? e : et;     // softmax numerator for the target class
        }
        float p = wl[ts] * (et / s);
        p = fminf(fmaxf(p, 1e-7f), 1.0f - 1e-7f);
        const float lossv = -(1.0f - p) * __logf(p);   // gamma == 1
        sumL += valid ? lossv : 0.0f;
        sumV += valid ? 1.0f  : 0.0f;
    }

    // wave32 butterfly reduction
#pragma unroll
    for (int off = 16; off > 0; off >>= 1) {
        sumL += __shfl_xor(sumL, off, 32);
        sumV += __shfl_xor(sumV, off, 32);
    }
    if ((tid & 31) == 0) {               // one leader per wave -> ds_add_f32
        atomicAdd(&red[0], sumL);
        atomicAdd(&red[1], sumV);
    }
    __syncthreads();
    if (tid == 0) {                      // striped global accumulation
        const int cell = blockIdx.x & (CELLS - 1);
        atomicAdd(&ws[cell],         red[0]);
        atomicAdd(&ws[CELLS + cell], red[1]);
    }
}

__global__ void finalize_kernel(const float* __restrict__ ws, float* __restrict__ out) {
    const int t = threadIdx.x;           // 32 threads, one wave
    float a = ws[t]         + ws[t + 32];
    float b = ws[CELLS + t] + ws[CELLS + t + 32];
#pragma unroll
    for (int off = 16; off > 0; off >>= 1) {
        a += __shfl_xor(a, off, 32);
        b += __shfl_xor(b, off, 32);
    }
    if (t == 0) out[0] = a / fmaxf(b, 1.0f);
}

extern "C" void kernel_launch(void* const* d_in, const int* in_sizes, int n_in,
                              void* d_out, int out_size, void* d_ws, size_t ws_size,
                              hipStream_t stream) {
    const float*     in  = (const float*)d_in[0];
    const long long* tgt = (const long long*)d_in[1];
    const float*     w   = (const float*)d_in[2];
    float* ws  = (float*)d_ws;
    float* out = (float*)d_out;

    zero_ws_kernel<<<dim3(1), dim3(2 * CELLS), 0, stream>>>(ws);
    focal_loss_kernel<<<dim3(GRID), dim3(BLOCK), 0, stream>>>(in, tgt, w, ws);
    finalize_kernel<<<dim3(1), dim3(32), 0, stream>>>(ws, out);
}